// TDGSPooling2d_17703855194494
// MI455X (gfx1250) — compile-verified
//
#include <hip/hip_runtime.h>
#include <stdint.h>

// Problem constants (x: 32x128x112x112 f32, temperature: 128x56x56 f32, K=2)
#define TDG_THREADS        128       // 4 wave32 waves per block
#define UNITS_PER_HCHUNK   16u       // row-pair units per half-chunk
#define UNIT_FLOATS        224u      // 2 input rows * 112 floats (contiguous)
#define UNIT_BYTES         896u
#define HCHUNK_FLOATS      (UNITS_PER_HCHUNK * UNIT_FLOATS)   // 3584 floats
#define HCHUNK_BYTES       (HCHUNK_FLOATS * 4u)               // 14336 B
#define CHUNKS_PER_BLOCK   8
#define TOTAL_HALF_UNITS   114688u   // (32*128*56)/2 row-pair units per half
#define HALF_UNIT_BYTEOFF  (TOTAL_HALF_UNITS * UNIT_BYTES)    // 102,760,448 B
#define HALF_ELEM_OFFSET   25690112u // same, in f32 elements (TDM row stride)
#define HALF_OUT_OFFSET    6422528u  // outputs for batches 16..31
#define N2_SPLIT           25690112u // threefry counter split (total_noise/2)
#define TEMP_ELEMS         401408u   // 128*56*56
#define W_IN               112u
#define WO_OUT             56u

#if __has_builtin(__builtin_amdgcn_tensor_load_to_lds)
#define USE_TDM 1
typedef uint32_t tdm_u32x4 __attribute__((ext_vector_type(4)));
typedef int      tdm_i32x8 __attribute__((ext_vector_type(8)));
typedef int      tdm_i32x4 __attribute__((ext_vector_type(4)));
#else
#define USE_TDM 0
#endif

#if __has_builtin(__builtin_amdgcn_s_wait_asynccnt)
#define WAIT_ASYNC(n) __builtin_amdgcn_s_wait_asynccnt(n)
#else
#define WAIT_ASYNC(n) asm volatile("s_wait_asynccnt " #n ::: "memory")
#endif

// JAX threefry2x32, key = (0, 42)  (jax.random.key(42)), 20 rounds.
__device__ __forceinline__ void threefry2x32_k42(uint32_t x0, uint32_t x1,
                                                 uint32_t& o0, uint32_t& o1) {
  const uint32_t ks0 = 0u;
  const uint32_t ks1 = 42u;
  const uint32_t ks2 = 0x1BD11BDAu ^ 42u;
  x0 += ks0; x1 += ks1;
#define TFR(r) { x0 += x1; x1 = __builtin_rotateleft32(x1, (r)); x1 ^= x0; }
  TFR(13) TFR(15) TFR(26) TFR(6)   x0 += ks1; x1 += ks2 + 1u;
  TFR(17) TFR(29) TFR(16) TFR(24)  x0 += ks2; x1 += ks0 + 2u;
  TFR(13) TFR(15) TFR(26) TFR(6)   x0 += ks0; x1 += ks1 + 3u;
  TFR(17) TFR(29) TFR(16) TFR(24)  x0 += ks1; x1 += ks2 + 4u;
  TFR(13) TFR(15) TFR(26) TFR(6)   x0 += ks2; x1 += ks0 + 5u;
#undef TFR
  o0 = x0; o1 = x1;
}

// u = (bits>>9 | 1.0f) - 1 + 1e-20 (JAX uniform(1e-20,1)); g = -ln(-ln u)
// via two v_log_f32: g = -ln(ln2) - ln2*log2(-log2(u))
__device__ __forceinline__ float gumbel_from_bits(uint32_t bits) {
  float f = __uint_as_float((bits >> 9) | 0x3F800000u) - 1.0f;
  float u = f + 1.0e-20f;
  float L = __log2f(u);                       // < 0
  return fmaf(-0.69314718055994531f, __log2f(-L), 0.36651292058166435f);
}

__global__ __launch_bounds__(TDG_THREADS)
void tdgs_pool2x2_kernel(const float* __restrict__ x,
                         const float* __restrict__ temperature,
                         float* __restrict__ out) {
  // LDS: [parity 0|1][half 0|1] * 14336 B, double-buffered staging
  __shared__ __align__(16) float smem[4 * HCHUNK_FLOATS];   // 57,344 B
  const unsigned tid = threadIdx.x;
  const unsigned q0 = blockIdx.x * CHUNKS_PER_BLOCK;
  const uint64_t xbase = (uint64_t)(uintptr_t)x;
  // low 32 bits of flat LDS address == LDS byte offset
  const unsigned lds_base = (unsigned)(uintptr_t)(&smem[0]);

#if USE_TDM
  // One Tensor Data Mover descriptor moves a whole chunk: a 2D tile of
  // 2 rows x 14336 B (both batch halves; row stride = half-tensor offset),
  // landing contiguously in the [p][h] double buffer. Issued by wave 0 only
  // (TDM ignores EXEC; a per-wave branch avoids 4x duplicate DMA).
  auto stage = [&](unsigned q, unsigned p) {
    const uint64_t ga = xbase + (uint64_t)q * HCHUNK_BYTES;
    tdm_u32x4 g0;
    g0.x = 1u;                                    // count=1, no gather/restore
    g0.y = lds_base + p * (2u * HCHUNK_BYTES);    // lds_addr (bytes)
    g0.z = (uint32_t)ga;                          // global_addr[31:0]
    g0.w = (uint32_t)(ga >> 32) | 0x80000000u;    // global_addr[56:32]|type=2
    tdm_i32x8 g1;
    g1[0] = (int)(2u << 16);                      // data_size=4B; no multicast
    g1[1] = (int)(HCHUNK_FLOATS << 16);           // tensor_dim0 lo
    g1[2] = (int)(2u << 16);                      // tensor_dim0 hi | tensor_dim1=2
    g1[3] = (int)(HCHUNK_FLOATS << 16);           // tensor_dim1 hi | tile_dim0
    g1[4] = 2;                                    // tile_dim1=2, tile_dim2=0
    g1[5] = (int)HALF_ELEM_OFFSET;                // tensor_dim0_stride lo
    g1[6] = 0;                                    // stride0 hi | stride1 lo
    g1[7] = 0;
    tdm_i32x4 g2 = {0, 0, 0, 0};                  // dims >=2 unused
    tdm_i32x4 g3 = {0, 0, 0, 0};
    tdm_i32x8 g4 = {0, 0, 0, 0, 0, 0, 0, 0};      // unused trailing group
    __builtin_amdgcn_tensor_load_to_lds(g0, g1, g2, g3, g4, 0);
  };
#else
  // Fallback: per-lane CDNA5 async global->LDS b128 staging (ASYNCcnt).
  auto stage = [&](unsigned q, unsigned p) {
    const unsigned g0 = q * HCHUNK_BYTES;
#pragma unroll
    for (int h = 0; h < 2; ++h) {
      const unsigned gb = g0 + (h ? HALF_UNIT_BYTEOFF : 0u);
      const unsigned lb = lds_base + ((p << 1) + h) * HCHUNK_BYTES;
#pragma unroll
      for (int j = 0; j < 7; ++j) {
        const unsigned s16 = (tid + (unsigned)j * TDG_THREADS) * 16u;
        const unsigned lds_addr = lb + s16;
        const unsigned gaddr = gb + s16;
        asm volatile("global_load_async_to_lds_b128 %0, %1, %2"
                     :: "v"(lds_addr), "v"(gaddr), "s"(xbase)
                     : "memory");
      }
    }
  };
#endif

  auto compute = [&](unsigned q, unsigned p) {
    const float* __restrict__ buf0 = smem + ((p << 1) + 0) * HCHUNK_FLOATS;
    const float* __restrict__ buf1 = smem + ((p << 1) + 1) * HCHUNK_FLOATS;
#pragma unroll 1
    for (int j = 0; j < 7; ++j) {
      const unsigned o = tid + (unsigned)j * TDG_THREADS;      // 0..895
      const unsigned ul = o / WO_OUT;
      const unsigned wo = o - ul * WO_OUT;
      const unsigned fo = ul * UNIT_FLOATS + 2u * wo;
      // 2x2 patch, row-major within patch: [a.x a.y b.x b.y]
      const float2 a0 = *reinterpret_cast<const float2*>(buf0 + fo);
      const float2 b0 = *reinterpret_cast<const float2*>(buf0 + fo + W_IN);
      const float2 a1 = *reinterpret_cast<const float2*>(buf1 + fo);
      const float2 b1 = *reinterpret_cast<const float2*>(buf1 + fo + W_IN);

      const unsigned oidx0 = (q * UNITS_PER_HCHUNK + ul) * WO_OUT + wo;
      const unsigned oidx1 = oidx0 + HALF_OUT_OFFSET;
      // batches b and b+16 share (c,ho,wo) -> one temperature load for both
      const float T = temperature[oidx0 % TEMP_ELEMS];
      const float inv_t = 1.0f / (fmaxf(T, 0.0f) + 0.1f);

      // threefry counter pairs (f, f+N2) feed BOTH paired outputs
      const uint32_t fbase = oidx0 * 4u;
      float g0v[4], g1v[4];
#pragma unroll
      for (int k = 0; k < 4; ++k) {
        uint32_t r0, r1;
        threefry2x32_k42(fbase + (uint32_t)k, fbase + (uint32_t)k + N2_SPLIT,
                         r0, r1);
        g0v[k] = gumbel_from_bits(r0);
        g1v[k] = gumbel_from_bits(r1);
      }
      const float p0[4] = {a0.x, a0.y, b0.x, b0.y};
      const float p1[4] = {a1.x, a1.y, b1.x, b1.y};
      float bs0 = fmaf(p0[0], inv_t, g0v[0]), bv0 = p0[0];
      float bs1 = fmaf(p1[0], inv_t, g1v[0]), bv1 = p1[0];
#pragma unroll
      for (int k = 1; k < 4; ++k) {
        const float s0 = fmaf(p0[k], inv_t, g0v[k]);
        if (s0 > bs0) { bs0 = s0; bv0 = p0[k]; }   // strict > = first argmax
        const float s1 = fmaf(p1[k], inv_t, g1v[k]);
        if (s1 > bs1) { bs1 = s1; bv1 = p1[k]; }
      }
      // streaming output: non-temporal, keep L2 for temperature reuse
      __builtin_nontemporal_store(bv0, out + oidx0);
      __builtin_nontemporal_store(bv1, out + oidx1);
    }
  };

  // Double-buffered pipeline: stage chunk i+1 while computing chunk i.
#if USE_TDM
  const bool wave0 = (tid < 32u);        // wave-uniform branch
  if (wave0) stage(q0, 0u);
#pragma unroll 1
  for (int i = 0; i < CHUNKS_PER_BLOCK; ++i) {
    const unsigned p = (unsigned)i & 1u;
    if (wave0) {
      if (i + 1 < CHUNKS_PER_BLOCK) {
        stage(q0 + (unsigned)i + 1u, p ^ 1u);
        __builtin_amdgcn_s_wait_tensorcnt(1);  // chunk i done; i+1 in flight
      } else {
        __builtin_amdgcn_s_wait_tensorcnt(0);
      }
    }
    __syncthreads();      // TDM LDS writes visible to all waves
    compute(q0 + (unsigned)i, p);
    __syncthreads();      // all waves done reading before buffer is reused
  }
#else
  stage(q0, 0u);
#pragma unroll 1
  for (int i = 0; i < CHUNKS_PER_BLOCK; ++i) {
    const unsigned p = (unsigned)i & 1u;
    if (i + 1 < CHUNKS_PER_BLOCK) {
      stage(q0 + (unsigned)i + 1u, p ^ 1u);
      WAIT_ASYNC(14);     // oldest 14 (chunk i) done; chunk i+1 in flight
    } else {
      WAIT_ASYNC(0);
    }
    __syncthreads();
    compute(q0 + (unsigned)i, p);
    __syncthreads();
  }
#endif
}

extern "C" void kernel_launch(void* const* d_in, const int* in_sizes, int n_in,
                              void* d_out, int out_size, void* d_ws, size_t ws_size,
                              hipStream_t stream) {
  (void)in_sizes; (void)n_in; (void)d_ws; (void)ws_size; (void)out_size;
  const float* x = (const float*)d_in[0];
  const float* temperature = (const float*)d_in[1];
  float* out = (float*)d_out;
  // 114688 half-units / 16 per chunk / 8 chunks per block = 896 blocks
  dim3 grid(TOTAL_HALF_UNITS / UNITS_PER_HCHUNK / CHUNKS_PER_BLOCK);
  dim3 block(TDG_THREADS);
  tdgs_pool2x2_kernel<<<grid, block, 0, stream>>>(x, temperature, out);
}